// DyHuCoG_28037546508449
// MI455X (gfx1250) — compile-verified
//
#include <hip/hip_runtime.h>
#include <hip/hip_bf16.h>
#include <math.h>

#define D 64
#define NUM_USERS_CONST 100000

typedef __attribute__((ext_vector_type(2))) float v2f;
typedef __attribute__((ext_vector_type(8))) float v8f;

// ---------- float <-> orderable-uint encoding for atomicMax on float ----------
__device__ __forceinline__ unsigned fenc(float f) {
    unsigned u = __float_as_uint(f);
    return (u >> 31) ? ~u : (u | 0x80000000u);
}
__device__ __forceinline__ float fdec(unsigned k) {
    return (k >> 31) ? __uint_as_float(k & 0x7FFFFFFFu) : __uint_as_float(~k);
}

// ---------------- degree: deg[row[e]] += 1 ----------------
__global__ void k_deg(const int* __restrict__ rows, float* __restrict__ deg, int E) {
    int e = blockIdx.x * blockDim.x + threadIdx.x;
    if (e < E) atomicAdd(&deg[rows[e]], 1.0f);
}

// ---------------- deg_inv = clip(deg,1)^-0.5 ----------------
__global__ void k_deginv(const float* __restrict__ deg, float* __restrict__ dinv, int N) {
    int i = blockIdx.x * blockDim.x + threadIdx.x;
    if (i < N) dinv[i] = rsqrtf(fmaxf(deg[i], 1.0f));
}

// ---------------- norm[e] = dinv[row]*dinv[col] ----------------
__global__ void k_norm(const int* __restrict__ rows, const int* __restrict__ cols,
                       const float* __restrict__ dinv, float* __restrict__ norm, int E) {
    int e = blockIdx.x * blockDim.x + threadIdx.x;
    if (e < E) norm[e] = dinv[rows[e]] * dinv[cols[e]];
}

// ---------------- propagate: xout[col] += norm * xin[row]; 64 threads/edge ----
__global__ __launch_bounds__(256) void k_prop(const float* __restrict__ xin,
                                              float* __restrict__ xout,
                                              const int* __restrict__ rows,
                                              const int* __restrict__ cols,
                                              const float* __restrict__ norm, int E) {
    long long t = (long long)blockIdx.x * blockDim.x + threadIdx.x;
    int e = (int)(t >> 6);
    int d = (int)(t & 63);
    if (e >= E) return;
    int r = rows[e];
    int c = cols[e];
    float nr = norm[e];
    atomicAdd(&xout[(size_t)c * D + d], nr * xin[(size_t)r * D + d]);
}

// ---------------- cnt[users[b]] += 1 ----------------
__global__ void k_cnt(const int* __restrict__ users, float* __restrict__ cnt, int B) {
    int b = blockIdx.x * blockDim.x + threadIdx.x;
    if (b < B) atomicAdd(&cnt[users[b]], 1.0f);
}

// ---------------- context[d] = mean_b x[users[b]][d] (single block, 64 thr) ----
__global__ void k_ctx(const float* __restrict__ x, const int* __restrict__ users,
                      float* __restrict__ ctx, int B) {
    int d = threadIdx.x;  // 0..63
    float s = 0.f;
    for (int b = 0; b < B; ++b) s += x[(size_t)users[b] * D + d];
    ctx[d] = s / (float)B;
}

// ---------------- pass 1: max over selected logits ----------------
__global__ __launch_bounds__(256) void k_max(const int* __restrict__ rows,
                                             const int* __restrict__ cols,
                                             const float* __restrict__ cnt,
                                             const float* __restrict__ x,
                                             const float* __restrict__ ctx,
                                             unsigned* __restrict__ maxkey, int E) {
    __shared__ float s_ctx[D];
    int tid = threadIdx.x;
    if (tid < D) s_ctx[tid] = ctx[tid];
    __syncthreads();
    int e = blockIdx.x * blockDim.x + tid;
    float v = -3.4e38f;
    if (e < E) {
        float mult = cnt[rows[e]];
        if (mult > 0.f) {
            const float* nr = x + (size_t)cols[e] * D;
            float dot = 0.f;
#pragma unroll
            for (int d = 0; d < D; ++d) dot += nr[d] * s_ctx[d];
            v = dot;
        }
    }
#pragma unroll
    for (int o = 16; o > 0; o >>= 1) v = fmaxf(v, __shfl_down(v, o, 32));
    if ((tid & 31) == 0 && v > -3.0e38f) atomicMax(maxkey, fenc(v));
}

// ---------------- pass 2: z and sum(w * neigh) ----------------
__global__ __launch_bounds__(256) void k_wsum(const int* __restrict__ rows,
                                              const int* __restrict__ cols,
                                              const float* __restrict__ cnt,
                                              const float* __restrict__ x,
                                              const float* __restrict__ ctx,
                                              const unsigned* __restrict__ maxkey,
                                              float* __restrict__ aggr,
                                              float* __restrict__ zp, int E) {
    __shared__ float s_ctx[D];
    __shared__ float s_acc[D];
    __shared__ float s_z;
    int tid = threadIdx.x;
    if (tid < D) { s_ctx[tid] = ctx[tid]; s_acc[tid] = 0.f; }
    if (tid == 0) s_z = 0.f;
    __syncthreads();
    int e = blockIdx.x * blockDim.x + tid;
    if (e < E) {
        float mult = cnt[rows[e]];
        if (mult > 0.f) {
            unsigned k = *maxkey;
            float mx = (k == 0u) ? 0.f : fdec(k);
            const float* nr = x + (size_t)cols[e] * D;
            float dot = 0.f;
#pragma unroll
            for (int d = 0; d < D; ++d) dot += nr[d] * s_ctx[d];
            float wgt = mult * expf(dot - mx);
            atomicAdd(&s_z, wgt);
#pragma unroll
            for (int d = 0; d < D; ++d) atomicAdd(&s_acc[d], wgt * nr[d]);
        }
    }
    __syncthreads();
    if (tid < D && s_acc[tid] != 0.f) atomicAdd(&aggr[tid], s_acc[tid]);
    if (tid == 0 && s_z != 0.f) atomicAdd(zp, s_z);
}

// ---------------- final scores via WMMA f32 16x16x4 ----------------
// D_tile = U(16x64) @ (I + aggr)^T(64x16); diag(D_tile) = scores; sigmoid.
// One wave (32 lanes) per 16-row batch tile. EXEC all ones, no divergence
// before the WMMA region (required by ISA).
__global__ __launch_bounds__(32) void k_score(const float* __restrict__ x,
                                              const int* __restrict__ users,
                                              const int* __restrict__ items,
                                              const float* __restrict__ aggr,
                                              const float* __restrict__ zp,
                                              float* __restrict__ out, int numUsers) {
    int lane = threadIdx.x;           // 0..31
    int tile = blockIdx.x;            // B/16 tiles
    int m = lane & 15;                // A-row index / B-col index for this lane
    int uIdx = users[tile * 16 + m];
    int iIdx = items[tile * 16 + m] + numUsers;
    const float* urow = x + (size_t)uIdx * D;
    const float* irow = x + (size_t)iIdx * D;
    float zi = 1.0f / fmaxf(zp[0], 1e-12f);

#if __has_builtin(__builtin_amdgcn_wmma_f32_16x16x4_f32)
    __shared__ float sD[256];
    // A 16x4 layout: lanes 0-15 -> K={0,1}; lanes 16-31 -> K={2,3}
    int kb = (lane < 16) ? 0 : 2;
    v8f c = {};
#pragma unroll
    for (int kk = 0; kk < D; kk += 4) {
        int k0 = kk + kb;
        v2f a;
        a.x = urow[k0];
        a.y = urow[k0 + 1];
        v2f b;
        b.x = irow[k0] + aggr[k0] * zi;
        b.y = irow[k0 + 1] + aggr[k0 + 1] * zi;
        c = __builtin_amdgcn_wmma_f32_16x16x4_f32(
            /*neg_a=*/false, a, /*neg_b=*/false, b,
            /*c_mod=*/(short)0, c, /*reuse_a=*/false, /*reuse_b=*/false);
    }
    // C/D layout: lanes 0-15: VGPR r -> M=r, N=lane; lanes 16-31: M=r+8, N=lane-16
    int Mbase = (lane < 16) ? 0 : 8;
#pragma unroll
    for (int r = 0; r < 8; ++r) sD[(Mbase + r) * 16 + m] = c[r];
    __syncthreads();
    if (lane < 16) {
        float sc = sD[lane * 16 + lane];
        out[tile * 16 + lane] = 1.0f / (1.0f + expf(-sc));
    }
#else
    if (lane < 16) {
        float sc = 0.f;
        for (int k = 0; k < D; ++k) sc += urow[k] * (irow[k] + aggr[k] * zi);
        out[tile * 16 + lane] = 1.0f / (1.0f + expf(-sc));
    }
#endif
}

extern "C" void kernel_launch(void* const* d_in, const int* in_sizes, int n_in,
                              void* d_out, int out_size, void* d_ws, size_t ws_size,
                              hipStream_t stream) {
    const float* embed = (const float*)d_in[0];
    const int* edge = (const int*)d_in[1];
    const int* users = (const int*)d_in[2];
    const int* items = (const int*)d_in[3];
    float* out = (float*)d_out;

    const int N = in_sizes[0] / D;      // 150000
    const int E = in_sizes[1] / 2;      // 2000000
    const int B = in_sizes[2];          // 1024
    const int* rows = edge;
    const int* cols = edge + E;

    // ---- workspace carve-out (256B aligned) ----
    char* w = (char*)d_ws;
    size_t off = 0;
    auto take = [&](size_t bytes) -> char* {
        char* p = w + off;
        off = (off + bytes + 255) & ~(size_t)255;
        return p;
    };
    float* xa    = (float*)take((size_t)N * D * sizeof(float));
    float* xb    = (float*)take((size_t)N * D * sizeof(float));
    float* deg   = (float*)take((size_t)N * sizeof(float));
    float* dinv  = (float*)take((size_t)N * sizeof(float));
    float* cnt   = (float*)take((size_t)N * sizeof(float));
    float* norm  = (float*)take((size_t)E * sizeof(float));
    float* ctx   = (float*)take(D * sizeof(float));
    float* aggr  = (float*)take(D * sizeof(float));
    float* scal  = (float*)take(256);           // [0]=z, [1]=maxkey(uint)
    float* zp = scal;
    unsigned* maxkey = (unsigned*)(scal + 1);

    const int T = 256;
    dim3 blkE((E + T - 1) / T);
    dim3 blkN((N + T - 1) / T);
    dim3 blkProp((unsigned)(((size_t)E * D + T - 1) / T));

    // ---- normalization ----
    hipMemsetAsync(deg, 0, (size_t)N * sizeof(float), stream);
    hipMemsetAsync(cnt, 0, (size_t)N * sizeof(float), stream);
    hipMemsetAsync(aggr, 0, D * sizeof(float), stream);
    hipMemsetAsync(scal, 0, 256, stream);
    k_deg<<<blkE, T, 0, stream>>>(rows, deg, E);
    k_deginv<<<blkN, T, 0, stream>>>(deg, dinv, N);
    k_norm<<<blkE, T, 0, stream>>>(rows, cols, dinv, norm, E);

    // ---- 3 propagation layers (ping-pong, zero target first) ----
    hipMemsetAsync(xa, 0, (size_t)N * D * sizeof(float), stream);
    k_prop<<<blkProp, T, 0, stream>>>(embed, xa, rows, cols, norm, E);
    hipMemsetAsync(xb, 0, (size_t)N * D * sizeof(float), stream);
    k_prop<<<blkProp, T, 0, stream>>>(xa, xb, rows, cols, norm, E);
    hipMemsetAsync(xa, 0, (size_t)N * D * sizeof(float), stream);
    k_prop<<<blkProp, T, 0, stream>>>(xb, xa, rows, cols, norm, E);
    // final embeddings in xa

    // ---- batch context / selection counts ----
    k_cnt<<<dim3((B + T - 1) / T), T, 0, stream>>>(users, cnt, B);
    k_ctx<<<1, D, 0, stream>>>(xa, users, ctx, B);

    // ---- softmax-weighted neighbor aggregation over selected edges ----
    k_max<<<blkE, T, 0, stream>>>(rows, cols, cnt, xa, ctx, maxkey, E);
    k_wsum<<<blkE, T, 0, stream>>>(rows, cols, cnt, xa, ctx, maxkey, aggr, zp, E);

    // ---- scores + sigmoid via WMMA (one wave per 16-row tile) ----
    k_score<<<dim3(B / 16), 32, 0, stream>>>(xa, users, items, aggr, zp, out,
                                             NUM_USERS_CONST);
}